// Envelop_90795608637681
// MI455X (gfx1250) — compile-verified
//
#include <hip/hip_runtime.h>
#include <hip/hip_bf16.h>
#include <math.h>

typedef __attribute__((ext_vector_type(2))) float v2f;
typedef __attribute__((ext_vector_type(8))) float v8f;

#define NUM_DET  16
#define NUM_ELEC 32
#define NUM_NUC  10
#define BATCH    1024
#define CLIPV    10000.0f

// One wave (32 lanes) per (b, k) determinant.
// LDS: 32x33 padded matrix + sigma/pi staging.
__global__ __launch_bounds__(32)
void envelope_det_kernel(const float* __restrict__ feat,     // (B, 32, 10)
                         const float* __restrict__ preDet,   // (B, 32, 512)
                         const float* __restrict__ piP,      // (16, 32, 10)
                         const float* __restrict__ sigmaP,   // (16, 32, 10)
                         float* __restrict__ detOut)         // (B, 16): sign*exp(logdet)
{
    __shared__ float M[NUM_ELEC][NUM_ELEC + 1];
    __shared__ float sSig[NUM_ELEC][NUM_NUC];
    __shared__ float sPi [NUM_ELEC][NUM_NUC];

    const int bk   = blockIdx.x;
    const int b    = bk >> 4;
    const int k    = bk & 15;
    const int lane = threadIdx.x;

    // --- stage sigma[k], pi[k] to LDS (320 floats each) ---
    for (int t = lane; t < NUM_ELEC * NUM_NUC; t += 32) {
        const int i = t / NUM_NUC, I = t % NUM_NUC;
        const int g = (k * NUM_ELEC + i) * NUM_NUC + I;
        sSig[i][I] = sigmaP[g];
        sPi [i][I] = piP[g];
    }
    // feat for this lane's column j = lane (contiguous 40B per lane)
    float f[NUM_NUC];
    {
        const size_t base = ((size_t)b * NUM_ELEC + lane) * NUM_NUC;
        #pragma unroll
        for (int I = 0; I < NUM_NUC; ++I) f[I] = feat[base + I];
    }
    __syncthreads();

    // --- build dets: lane owns column j = lane ---
    for (int i = 0; i < NUM_ELEC; ++i) {
        float om = 0.0f;
        #pragma unroll
        for (int I = 0; I < NUM_NUC; ++I)
            om += sPi[i][I] * __expf(-sSig[i][I] * f[I]);
        const float pre =
            preDet[((size_t)b * NUM_ELEC + i) * (NUM_DET * NUM_ELEC) + lane * NUM_DET + k];
        float v = pre * om;
        v = fminf(fmaxf(v, -CLIPV), CLIPV);
        M[i][lane] = v;
    }
    __syncthreads();

    // --- blocked LU, panel width 4, trailing update via V_WMMA_F32_16X16X4_F32 ---
    int swaps = 0;
    const int half = lane >> 4;   // 0 -> K pair {0,1}; 1 -> {2,3}
    const int l15  = lane & 15;

    for (int c0 = 0; c0 < NUM_ELEC; c0 += 4) {
        // ---- panel factorization (cols c0..c0+3, partial pivoting, full-row swaps) ----
        for (int cc = 0; cc < 4; ++cc) {
            const int c = c0 + cc;
            // argmax |M[r][c]| over r >= c (butterfly, wave32)
            float val = (lane >= c) ? fabsf(M[lane][c]) : -1.0f;
            int   idx = lane;
            #pragma unroll
            for (int off = 16; off > 0; off >>= 1) {
                const float ov = __shfl_xor(val, off);
                const int   oi = __shfl_xor(idx, off);
                if (ov > val || (ov == val && oi < idx)) { val = ov; idx = oi; }
            }
            if (idx != c) {
                ++swaps;
                const float a0 = M[c][lane];
                const float a1 = M[idx][lane];
                M[c][lane]   = a1;
                M[idx][lane] = a0;
            }
            __syncthreads();
            const float piv = M[c][c];
            if (lane > c) {                       // lane = row r
                const float m = M[lane][c] / piv;
                M[lane][c] = m;
                #pragma unroll
                for (int jj = cc + 1; jj < 4; ++jj)
                    M[lane][c0 + jj] -= m * M[c][c0 + jj];
            }
            __syncthreads();
        }
        if (c0 == NUM_ELEC - 4) break;

        // ---- U12 = L11^{-1} * A12 (forward substitution; lane = column j) ----
        if (lane >= c0 + 4) {
            #pragma unroll
            for (int rr = 1; rr < 4; ++rr) {
                float u = M[c0 + rr][lane];
                #pragma unroll
                for (int t = 0; t < rr; ++t)
                    u -= M[c0 + rr][c0 + t] * M[c0 + t][lane];
                M[c0 + rr][lane] = u;
            }
        }
        __syncthreads();

        // ---- trailing update: A22 -= L21 * U12 (k=4), four 16x16 WMMA tiles ----
        #pragma unroll
        for (int tm = 0; tm < NUM_ELEC; tm += 16) {
            #pragma unroll
            for (int tn = 0; tn < NUM_ELEC; tn += 16) {
                // A fragment: row = tm+l15, K = c0 + 2*half + {0,1}; negate (subtract)
                const int row   = tm + l15;
                const bool rOK  = (row >= c0 + 4);
                v2f a;
                a.x = rOK ? -M[row][c0 + 2 * half + 0] : 0.0f;
                a.y = rOK ? -M[row][c0 + 2 * half + 1] : 0.0f;
                // B fragment: col = tn+l15, K striped like A
                const int colB  = tn + l15;
                const bool cOK  = (colB >= c0 + 4);
                v2f bf;
                bf.x = cOK ? M[c0 + 2 * half + 0][colB] : 0.0f;
                bf.y = cOK ? M[c0 + 2 * half + 1][colB] : 0.0f;
                // C fragment (16x16 f32 layout): VGPR v -> row tm + v + 8*half
                const int rbase = tm + 8 * half;
                v8f cf;
                #pragma unroll
                for (int v = 0; v < 8; ++v) cf[v] = M[rbase + v][tn + l15];
                __syncthreads();
                v8f d = __builtin_amdgcn_wmma_f32_16x16x4_f32(
                    false, a, false, bf, (short)0, cf, false, false);
                #pragma unroll
                for (int v = 0; v < 8; ++v) M[rbase + v][tn + l15] = d[v];
                __syncthreads();
            }
        }
        __syncthreads();
    }

    // --- log|det| and sign from diagonal + permutation parity ---
    const float dval = M[lane][lane];
    float lsum = __logf(fabsf(dval));
    #pragma unroll
    for (int off = 16; off > 0; off >>= 1)
        lsum += __shfl_xor(lsum, off);
    const unsigned long long negmask = __ballot(dval < 0.0f);
    const int nneg = __popcll(negmask);
    const float sgn = ((nneg + swaps) & 1) ? -1.0f : 1.0f;
    if (lane == 0)
        detOut[bk] = sgn * __expf(lsum);
}

__global__ __launch_bounds__(256)
void reduce_logpsi_kernel(const float* __restrict__ detOut,  // (B, 16)
                          float* __restrict__ out)           // (B, 1)
{
    const int b = blockIdx.x * blockDim.x + threadIdx.x;
    if (b >= BATCH) return;
    float s = 0.0f;
    #pragma unroll
    for (int k = 0; k < NUM_DET; ++k) s += detOut[b * NUM_DET + k];
    out[b] = __logf(fabsf(s));
}

extern "C" void kernel_launch(void* const* d_in, const int* in_sizes, int n_in,
                              void* d_out, int out_size, void* d_ws, size_t ws_size,
                              hipStream_t stream) {
    const float* feat   = (const float*)d_in[0];  // (1024, 32, 10, 1)
    const float* preDet = (const float*)d_in[1];  // (1024, 32, 512)
    const float* piP    = (const float*)d_in[2];  // (16, 32, 10)
    const float* sigmaP = (const float*)d_in[3];  // (16, 32, 10)
    float* out    = (float*)d_out;                // (1024, 1)
    float* detWs  = (float*)d_ws;                 // 16384 floats

    envelope_det_kernel<<<dim3(BATCH * NUM_DET), dim3(32), 0, stream>>>(
        feat, preDet, piP, sigmaP, detWs);
    reduce_logpsi_kernel<<<dim3((BATCH + 255) / 256), dim3(256), 0, stream>>>(
        detWs, out);
}